// HungarianMatcher_721554506298
// MI455X (gfx1250) — compile-verified
//
#include <hip/hip_runtime.h>

// HungarianMatcher for MI455X (gfx1250, wave32).
// Phase 1: softmax + one-hot gather as f32 WMMA (V_WMMA_F32_16X16X4_F32).
// Phase 2: wave-parallel Jonker-Volgenant shortest augmenting path per batch.

typedef __attribute__((ext_vector_type(2))) float v2f;
typedef __attribute__((ext_vector_type(8))) float v8f;

#define B_   256
#define Q_   60
#define A_   128
#define P_   4
#define N_   16
#define QPAD 64
#define INFC 1000000000.0f

// ---------------------------------------------------------------------------
// Kernel A: build cost[b][n][q] = -(softmax_a(b,q)[ga[b,n]] + softmax_p(b,q)[gp[b,n]])
// One 64-thread block (2 waves) per batch. Gather realized as WMMA against a
// one-hot selection matrix (exact in f32).
// ---------------------------------------------------------------------------
__global__ void __launch_bounds__(64)
cost_kernel(const float* __restrict__ la, const float* __restrict__ lp,
            const int* __restrict__ ga, const int* __restrict__ gp,
            float* __restrict__ cost)
{
    __shared__ float Ea[QPAD * A_];   // normalized aspect probs, rows 60..63 zero
    __shared__ float Ep[QPAD * P_];   // normalized polarity probs
    __shared__ int   sga[N_], sgp[N_];

    const int b = blockIdx.x;
    const int t = threadIdx.x;

    if (t < N_) { sga[t] = ga[b * N_ + t]; sgp[t] = gp[b * N_ + t]; }

    // Coalesced staging of logits into LDS (zero the padded rows).
    for (int idx = t; idx < QPAD * A_; idx += 64)
        Ea[idx] = (idx < Q_ * A_) ? la[(size_t)b * Q_ * A_ + idx] : 0.0f;
    for (int idx = t; idx < QPAD * P_; idx += 64)
        Ep[idx] = (idx < Q_ * P_) ? lp[(size_t)b * Q_ * P_ + idx] : 0.0f;
    __syncthreads();

    // Per-query softmax, pre-normalized in LDS so aspect+polarity WMMAs can
    // share one accumulator chain.
    if (t < Q_) {
        const int q = t;
        float mx = -INFC;
        for (int k = 0; k < A_; ++k) mx = fmaxf(mx, Ea[q * A_ + k]);
        float s = 0.0f;
        for (int k = 0; k < A_; ++k) { float e = expf(Ea[q * A_ + k] - mx); Ea[q * A_ + k] = e; s += e; }
        const float r = 1.0f / s;
        for (int k = 0; k < A_; ++k) Ea[q * A_ + k] *= r;

        float mp = fmaxf(fmaxf(Ep[q * P_ + 0], Ep[q * P_ + 1]),
                         fmaxf(Ep[q * P_ + 2], Ep[q * P_ + 3]));
        float sp = 0.0f;
        for (int k = 0; k < P_; ++k) { float e = expf(Ep[q * P_ + k] - mp); Ep[q * P_ + k] = e; sp += e; }
        const float rp = 1.0f / sp;
        for (int k = 0; k < P_; ++k) Ep[q * P_ + k] *= rp;
    }
    __syncthreads();

    // WMMA gather: D[qm][n] = sum_k E[qm][k] * onehot(gold[n])[k].
    // A-matrix 16x4 f32 layout: lanes 0-15 -> M, VGPR0/1 = K0/K1; lanes 16-31 = K2/K3.
    // B-matrix 4x16 f32 layout: column N = lane&15; lanes 0-15 hold K0/K1, lanes 16-31 K2/K3.
    const int wave = t >> 5;
    const int l    = t & 31;
    const int mn   = l & 15;   // M row of A-tile and N col of B
    const int kh   = l >> 4;   // K-half

    for (int tile = wave; tile < 4; tile += 2) {
        v8f c;
#pragma unroll
        for (int r = 0; r < 8; ++r) c[r] = 0.0f;

        const int qrow = tile * 16 + mn;
        const int gaN  = sga[mn];
        const int gpN  = sgp[mn];

#pragma unroll 4
        for (int kk = 0; kk < A_ / 4; ++kk) {
            const int k0 = 4 * kk + 2 * kh;
            v2f a;  a.x  = Ea[qrow * A_ + k0];     a.y  = Ea[qrow * A_ + k0 + 1];
            v2f bm; bm.x = (gaN == k0) ? 1.0f : 0.0f;
                    bm.y = (gaN == k0 + 1) ? 1.0f : 0.0f;
            c = __builtin_amdgcn_wmma_f32_16x16x4_f32(false, a, false, bm,
                                                      (short)0, c, false, false);
        }
        {   // polarity: K = 4 exactly -> single WMMA
            const int k0 = 2 * kh;
            v2f a;  a.x  = Ep[qrow * P_ + k0];     a.y  = Ep[qrow * P_ + k0 + 1];
            v2f bm; bm.x = (gpN == k0) ? 1.0f : 0.0f;
                    bm.y = (gpN == k0 + 1) ? 1.0f : 0.0f;
            c = __builtin_amdgcn_wmma_f32_16x16x4_f32(false, a, false, bm,
                                                      (short)0, c, false, false);
        }

        // D layout: VGPR r -> M = r (lanes 0-15) or r+8 (lanes 16-31), N = lane&15.
#pragma unroll
        for (int r = 0; r < 8; ++r) {
            const int q = tile * 16 + r + ((l >= 16) ? 8 : 0);
            if (q < Q_)
                cost[((size_t)b * N_ + mn) * QPAD + q] = -c[r];
        }
    }
}

// ---------------------------------------------------------------------------
// Kernel B: Jonker-Volgenant per batch, one wave32 per block.
// Columns j = 0..60 (0 = dummy) spread 2 per lane; argmin via shfl_xor butterfly.
// ---------------------------------------------------------------------------
__global__ void __launch_bounds__(32)
hungarian_kernel(const float* __restrict__ cost,
                 const int* __restrict__ ga, const int* __restrict__ gp,
                 float* __restrict__ out)
{
    __shared__ float cl[N_ * QPAD];
    __shared__ float u[N_ + 1];
    __shared__ int   p[Q_ + 1];
    __shared__ int   way[Q_ + 1];
    __shared__ int   colrow[N_];

    const int b = blockIdx.x;
    const int l = threadIdx.x;

    for (int idx = l; idx < N_ * QPAD; idx += 32)
        cl[idx] = cost[(size_t)b * N_ * QPAD + idx];
    if (l <= N_) u[l] = 0.0f;
#pragma unroll
    for (int s = 0; s < 2; ++s) { const int j = l + 32 * s; if (j <= Q_) p[j] = 0; }
    if (l < N_) colrow[l] = -1;

    float v[2] = {0.0f, 0.0f};
    __syncthreads();

    for (int i = 0; i < N_; ++i) {
        float minv[2] = {INFC, INFC};
        bool  used[2] = {false, false};
        if (l == 0) p[0] = i + 1;
#pragma unroll
        for (int s = 0; s < 2; ++s) { const int j = l + 32 * s; if (j <= Q_) way[j] = 0; }
        int j0 = 0;
        __syncthreads();

        while (true) {
            const int i0 = p[j0];
            if (i0 == 0) break;
            const float ui0 = u[i0];

#pragma unroll
            for (int s = 0; s < 2; ++s) {
                const int j = l + 32 * s;
                if (j == j0) used[s] = true;
            }

            // update minv/way; gather masked argmin candidates
            float bv = 2.0f * INFC; int bj = 1 << 30;
#pragma unroll
            for (int s = 0; s < 2; ++s) {
                const int j = l + 32 * s;
                if (j >= 1 && j <= Q_) {
                    if (!used[s]) {
                        const float cur = cl[(i0 - 1) * QPAD + (j - 1)] - ui0 - v[s];
                        if (cur < minv[s]) { minv[s] = cur; way[j] = j0; }
                    }
                    const float cand = used[s] ? INFC : minv[s];
                    if (cand < bv || (cand == bv && j < bj)) { bv = cand; bj = j; }
                }
            }
            // wave-wide min with first-index tie-break (matches jnp.argmin)
#pragma unroll
            for (int off = 16; off > 0; off >>= 1) {
                const float ov = __shfl_xor(bv, off, 32);
                const int   oj = __shfl_xor(bj, off, 32);
                if (ov < bv || (ov == bv && oj < bj)) { bv = ov; bj = oj; }
            }
            const float delta = bv;

            // dual updates: used columns carry distinct rows -> distinct u slots
#pragma unroll
            for (int s = 0; s < 2; ++s) {
                const int j = l + 32 * s;
                if (j <= Q_) {
                    if (used[s]) { u[p[j]] += delta; v[s] -= delta; }
                    else if (j >= 1) minv[s] -= delta;
                }
            }
            j0 = bj;
            __syncthreads();
        }

        // augmenting path chase (serial, short)
        if (l == 0) {
            int j = j0;
            while (j != 0) { const int j1 = way[j]; p[j] = p[j1]; j = j1; }
        }
        __syncthreads();
    }

    // invert assignment: col_of_row
#pragma unroll
    for (int s = 0; s < 2; ++s) {
        const int j = l + 32 * s;
        if (j >= 1 && j <= Q_) { const int r = p[j]; if (r > 0) colrow[r - 1] = j - 1; }
    }
    __syncthreads();

    if (l < N_) {
        const int o = b * N_ + l;
        out[o]                 = (float)l;              // row_ind
        out[B_ * N_ + o]       = (float)colrow[l];      // col_ind
        out[2 * B_ * N_ + o]   = (float)ga[o];          // gold_aspects
        out[3 * B_ * N_ + o]   = (float)gp[o];          // gold_polarities
    }
}

extern "C" void kernel_launch(void* const* d_in, const int* in_sizes, int n_in,
                              void* d_out, int out_size, void* d_ws, size_t ws_size,
                              hipStream_t stream) {
    const float* la = (const float*)d_in[0];   // [256,60,128]
    const float* lp = (const float*)d_in[1];   // [256,60,4]
    const int*   ga = (const int*)d_in[2];     // [256,16]
    const int*   gp = (const int*)d_in[3];     // [256,16]
    float* out = (float*)d_out;                // 4 x [256,16] concatenated
    float* cost_ws = (float*)d_ws;             // [256][16][64] = 1 MB

    cost_kernel<<<B_, 64, 0, stream>>>(la, lp, ga, gp, cost_ws);
    hungarian_kernel<<<B_, 32, 0, stream>>>(cost_ws, ga, gp, out);
}